// GlobalContinualSVGP_11269994184838
// MI455X (gfx1250) — compile-verified
//
#include <hip/hip_runtime.h>
#include <math.h>

typedef __attribute__((ext_vector_type(2))) float v2f;
typedef __attribute__((ext_vector_type(8))) float v8f;

#define B_    4096
#define DIN_  256
#define OUT_  32
#define M_    128
#define NH_   2
#define JIT_  1.0e-4f

static_assert(NH_ == 2, "x2 kernel hardcodes NH=2");

// ---- workspace layout (float offsets) ----
constexpr size_t OFF_INV2 = 0;                       // NH*DIN = 512
constexpr size_t OFF_SF2  = OFF_INV2 + 512;          // 2 (pad to 8)
constexpr size_t OFF_Z2   = OFF_SF2  + 8;            // NH*OUT*M = 8192
constexpr size_t OFF_X2   = OFF_Z2   + 8192;         // NH*B = 8192
constexpr size_t OFF_R    = OFF_X2   + 8192;         // NH*OUT*M = 8192
constexpr size_t OFF_ZSC  = OFF_R    + 8192;         // NH*OUT*M*DIN = 2097152
constexpr size_t OFF_L    = OFF_ZSC  + 2097152;      // NH*OUT*M*M = 1048576
constexpr size_t OFF_LINV = OFF_L    + 1048576;
constexpr size_t OFF_Q    = OFF_LINV + 1048576;
constexpr size_t OFF_W    = OFF_Q    + 1048576;      // temp

// fp32 WMMA: D(16x16) = A(16x4) * B(4x16) + C, 8-arg VOP3P form
static __device__ __forceinline__ v8f wmma4(v2f a, v2f b, v8f c) {
  return __builtin_amdgcn_wmma_f32_16x16x4_f32(false, a, false, b, (short)0, c,
                                               false, false);
}

// ---------------- hyper: inv2[h][d] = exp(-2*theta[h][1+d]); sf2[h] ----------
__global__ void k_hyper(const float* __restrict__ theta,
                        float* __restrict__ inv2, float* __restrict__ sf2) {
  int t = threadIdx.x;
  if (t < NH_ * DIN_) {
    int h = t / DIN_, d = t % DIN_;
    inv2[t] = expf(-2.0f * theta[h * (DIN_ + 1) + 1 + d]);
  }
  if (t < NH_) sf2[t] = expf(theta[t * (DIN_ + 1)]);
}

// ---------------- zsc[h,o,m,d] = z*inv2 ; z2[h,o,m] = sum z*zsc ---------------
__global__ __launch_bounds__(128) void k_zsc(const float* __restrict__ z,
                                             const float* __restrict__ inv2,
                                             float* __restrict__ zsc,
                                             float* __restrict__ z2) {
  const int o = blockIdx.x, h = blockIdx.y, p = h * OUT_ + o;
  const int m = threadIdx.x;
  const float* zr = z + (size_t)(o * M_ + m) * DIN_;
  const float* iv = inv2 + (size_t)h * DIN_;
  float* dst = zsc + (size_t)(p * M_ + m) * DIN_;
  float acc = 0.0f;
  for (int d = 0; d < DIN_; ++d) {
    float s = zr[d] * iv[d];
    dst[d] = s;
    acc += zr[d] * s;
  }
  z2[p * M_ + m] = acc;
}

// ---------------- x2[h,b] = sum_d x^2 * inv2 ---------------------------------
__global__ __launch_bounds__(128) void k_x2(const float* __restrict__ x,
                                            const float* __restrict__ inv2,
                                            float* __restrict__ x2) {
  const int b = blockIdx.x * 128 + threadIdx.x;
  const float* xr = x + (size_t)b * DIN_;
  float a0 = 0.0f, a1 = 0.0f;
  for (int d = 0; d < DIN_; ++d) {
    float v = xr[d] * xr[d];
    a0 += v * inv2[d];
    a1 += v * inv2[DIN_ + d];
  }
  x2[b] = a0;
  x2[B_ + b] = a1;
}

// ---------------- kuu (with jitter) into L buffer ----------------------------
__global__ __launch_bounds__(128) void k_kuu(const float* __restrict__ z,
                                             const float* __restrict__ zsc,
                                             const float* __restrict__ z2,
                                             const float* __restrict__ sf2,
                                             float* __restrict__ kuug) {
  const int o = blockIdx.x, h = blockIdx.y, p = h * OUT_ + o;
  const int m = threadIdx.x;
  const float* zm = zsc + (size_t)(p * M_ + m) * DIN_;
  const float z2m = z2[p * M_ + m];
  const float sf = sf2[h];
  float* dst = kuug + (size_t)p * M_ * M_ + (size_t)m * M_;
  for (int n = 0; n < M_; ++n) {
    const float* zn = z + (size_t)(o * M_ + n) * DIN_;
    float c = 0.0f;
    for (int d = 0; d < DIN_; ++d) c += zm[d] * zn[d];
    float d2 = fmaxf(z2m + z2[p * M_ + n] - 2.0f * c, 0.0f);
    dst[n] = sf * expf(-0.5f * d2) + ((m == n) ? JIT_ : 0.0f);
  }
}

// ---------------- in-place Cholesky (per (h,o), 128 threads) -----------------
__global__ __launch_bounds__(128) void k_chol(float* __restrict__ Lg) {
  __shared__ float A[M_ * M_];  // 64 KB
  const int p = blockIdx.x;
  const int tid = threadIdx.x;
  float* src = Lg + (size_t)p * M_ * M_;
  for (int idx = tid; idx < M_ * M_; idx += 128) A[idx] = src[idx];
  __syncthreads();
  for (int k = 0; k < M_; ++k) {
    if (tid == k) A[k * M_ + k] = sqrtf(A[k * M_ + k]);
    __syncthreads();
    float lkk = A[k * M_ + k];
    if (tid > k) A[tid * M_ + k] /= lkk;
    __syncthreads();
    if (tid > k) {
      float lik = A[tid * M_ + k];
      for (int j = k + 1; j <= tid; ++j) A[tid * M_ + j] -= lik * A[j * M_ + k];
    }
    __syncthreads();
  }
  for (int idx = tid; idx < M_ * M_; idx += 128) {
    int i = idx >> 7, j = idx & 127;
    src[idx] = (j <= i) ? A[idx] : 0.0f;
  }
}

// ---------------- Linv (column per thread), y = L^-1 u, r = Linv^T y ---------
__global__ __launch_bounds__(128) void k_linv(const float* __restrict__ Lg,
                                              const float* __restrict__ u_mean,
                                              float* __restrict__ Linvg,
                                              float* __restrict__ rg) {
  __shared__ float Lt[M_ * (M_ + 1) / 2];  // packed lower, ~33 KB
  __shared__ float yl[M_];
  const int p = blockIdx.x;
  const int o = p % OUT_;
  const int c = threadIdx.x;
  const float* src = Lg + (size_t)p * M_ * M_;
  for (int i = 0; i < M_; ++i) {
    int base = i * (i + 1) / 2;
    for (int j = c; j <= i; j += 128) Lt[base + j] = src[i * M_ + j];
  }
  __syncthreads();
  float* X = Linvg + (size_t)p * M_ * M_;  // column c written by thread c
  for (int i = 0; i < c; ++i) X[i * M_ + c] = 0.0f;
  for (int i = c; i < M_; ++i) {
    int base = i * (i + 1) / 2;
    float s = (i == c) ? 1.0f : 0.0f;
    for (int j = c; j < i; ++j) s -= Lt[base + j] * X[j * M_ + c];
    X[i * M_ + c] = s / Lt[base + i];
  }
  if (c == 0) {
    for (int i = 0; i < M_; ++i) {
      int base = i * (i + 1) / 2;
      float s = u_mean[o * M_ + i];
      for (int j = 0; j < i; ++j) s -= Lt[base + j] * yl[j];
      yl[i] = s / Lt[base + i];
    }
  }
  __syncthreads();
  float r = 0.0f;  // own column of X -> same-thread global RAW, ordered
  for (int i = c; i < M_; ++i) r += yl[i] * X[i * M_ + c];
  rg[p * M_ + c] = r;
}

// ---------------- W = Linv * tril(u_tril_vec) --------------------------------
__global__ __launch_bounds__(128) void k_w(const float* __restrict__ Linvg,
                                           const float* __restrict__ tril,
                                           float* __restrict__ Wg) {
  const int p = blockIdx.x, o = p % OUT_;
  const int m = threadIdx.x;
  const float* Li = Linvg + (size_t)p * M_ * M_ + (size_t)m * M_;
  const float* T = tril + (size_t)o * (M_ * (M_ + 1) / 2);
  float* Wrow = Wg + (size_t)p * M_ * M_ + (size_t)m * M_;
  for (int j = 0; j < M_; ++j) {
    float s = 0.0f;
    for (int i = j; i <= m; ++i) s += Li[i] * T[i * (i + 1) / 2 + j];
    Wrow[j] = s;  // j>m -> loop empty -> 0
  }
}

// ---------------- Q = W^T * Linv ---------------------------------------------
__global__ __launch_bounds__(128) void k_q(const float* __restrict__ Linvg,
                                           const float* __restrict__ Wg,
                                           float* __restrict__ Qg) {
  const int p = blockIdx.x;
  const int a = threadIdx.x;
  const float* Lb = Linvg + (size_t)p * M_ * M_;
  const float* Wb = Wg + (size_t)p * M_ * M_;
  float* Qrow = Qg + (size_t)p * M_ * M_ + (size_t)a * M_;
  for (int b = 0; b < M_; ++b) {
    float s = 0.0f;
    for (int m = a; m < M_; ++m) s += Wb[m * M_ + a] * Lb[m * M_ + b];
    Qrow[b] = s;
  }
}

// ---------------- fused main kernel: Kuf tile + two GEMMs + reductions -------
// grid (B/128, OUT, NH), 256 threads (8 waves). Wave w: stage1 owns m-strip
// [16w,16w+16); stage2 owns n-strip [16w,16w+16).
__global__ __launch_bounds__(256) void k_main(
    const float* __restrict__ x, const float* __restrict__ zsc,
    const float* __restrict__ z2g, const float* __restrict__ x2g,
    const float* __restrict__ sf2g, const float* __restrict__ linvg,
    const float* __restrict__ qg, const float* __restrict__ rg,
    float* __restrict__ out) {
  constexpr int S = 132;                 // padded LDS row stride (floats)
  __shared__ float kuf[M_ * S];          // 67.5 KB, (k=m rows) x (n cols)
  const int h = blockIdx.z, o = blockIdx.y, p = h * OUT_ + o;
  const int b0 = blockIdx.x * 128;
  const int tid = threadIdx.x;
  const int w = tid >> 5;
  const int lane = tid & 31;
  const int lo = lane & 15;
  const int hi = lane >> 4;

  // ---- stage 1: Kuf tile = sf2*exp(-0.5*d2), cross via fp32 WMMA ----
  // A frag: zsc row m=16w+lo, k pair at 2*hi     (16x4, 2 VGPRs)
  // B frag: raw x row b=b0+16t+lo, d pair at 2*hi (4x16, 2 VGPRs)
  const float* arow = zsc + (size_t)(p * M_ + (w << 4) + lo) * DIN_ + 2 * hi;
  const float* xrow = x + (size_t)(b0 + lo) * DIN_ + 2 * hi;
  v8f acc[8] = {};
  for (int k0 = 0; k0 < DIN_; k0 += 4) {
    v2f a = *(const v2f*)(arow + k0);
#pragma unroll
    for (int t = 0; t < 8; ++t) {
      v2f b = *(const v2f*)(xrow + (size_t)t * 16 * DIN_ + k0);
      acc[t] = wmma4(a, b, acc[t]);
    }
  }
  float z2v[8];
#pragma unroll
  for (int v = 0; v < 8; ++v)
    z2v[v] = z2g[p * M_ + (w << 4) + (hi << 3) + v];
  const float sf2v = sf2g[h];
#pragma unroll
  for (int t = 0; t < 8; ++t) {
    float x2v = x2g[(size_t)h * B_ + b0 + t * 16 + lo];
#pragma unroll
    for (int v = 0; v < 8; ++v) {
      float d2 = fmaxf(z2v[v] + x2v - 2.0f * acc[t][v], 0.0f);
      // D layout: m = v + 8*hi (within strip), n = lo (within tile t)
      kuf[((w << 4) + (hi << 3) + v) * S + t * 16 + lo] = sf2v * expf(-0.5f * d2);
    }
  }
  __syncthreads();

  // ---- stage 2: V = Linv*Kuf, U = Q*Kuf ; per-column sums of squares ----
  const float* lbase = linvg + (size_t)p * M_ * M_;
  const float* qbase = qg + (size_t)p * M_ * M_;
  const int ncol = (w << 4) + lo;  // this wave's column for this lane
  float d1 = 0.0f, d2s = 0.0f;
  for (int mt = 0; mt < 8; ++mt) {
    const float* la = lbase + (size_t)((mt << 4) + lo) * M_ + 2 * hi;
    const float* qa = qbase + (size_t)((mt << 4) + lo) * M_ + 2 * hi;
    v8f av = {};
    v8f au = {};
    for (int k0 = 0; k0 < M_; k0 += 4) {
      int kk = k0 + 2 * hi;
      v2f bf;
      bf.x = kuf[kk * S + ncol];
      bf.y = kuf[(kk + 1) * S + ncol];
      v2f af = *(const v2f*)(la + k0);
      av = wmma4(af, bf, av);
      v2f ag = *(const v2f*)(qa + k0);
      au = wmma4(ag, bf, au);
    }
#pragma unroll
    for (int v = 0; v < 8; ++v) {
      d1 += av[v] * av[v];
      d2s += au[v] * au[v];
    }
  }

  // mu_n = sum_m r[m]*Kuf[m][n]; split m by parity across lane halves
  const float* rv = rg + p * M_;
  float mu = 0.0f;
  for (int m = hi; m < M_; m += 2) mu += rv[m] * kuf[m * S + ncol];

  // lanes l and l^16 hold same n with complementary m halves
  mu += __shfl_xor(mu, 16, 32);
  d1 += __shfl_xor(d1, 16, 32);
  d2s += __shfl_xor(d2s, 16, 32);

  if (hi == 0) {
    size_t ob = (size_t)p * B_ + b0 + ncol;
    out[ob] = mu;
    out[(size_t)NH_ * OUT_ * B_ + ob] = sf2v - d1 + d2s;
  }
}

extern "C" void kernel_launch(void* const* d_in, const int* in_sizes, int n_in,
                              void* d_out, int out_size, void* d_ws,
                              size_t ws_size, hipStream_t stream) {
  const float* x = (const float*)d_in[0];
  const float* z = (const float*)d_in[1];
  const float* u_mean = (const float*)d_in[2];
  const float* u_tril = (const float*)d_in[3];
  const float* theta = (const float*)d_in[4];
  float* ws = (float*)d_ws;
  float* inv2 = ws + OFF_INV2;
  float* sf2 = ws + OFF_SF2;
  float* z2 = ws + OFF_Z2;
  float* x2 = ws + OFF_X2;
  float* rg = ws + OFF_R;
  float* zsc = ws + OFF_ZSC;
  float* Lb = ws + OFF_L;
  float* Linv = ws + OFF_LINV;
  float* Qb = ws + OFF_Q;
  float* Wb = ws + OFF_W;

  k_hyper<<<1, 512, 0, stream>>>(theta, inv2, sf2);
  k_zsc<<<dim3(OUT_, NH_), 128, 0, stream>>>(z, inv2, zsc, z2);
  k_x2<<<B_ / 128, 128, 0, stream>>>(x, inv2, x2);
  k_kuu<<<dim3(OUT_, NH_), 128, 0, stream>>>(z, zsc, z2, sf2, Lb);
  k_chol<<<NH_ * OUT_, 128, 0, stream>>>(Lb);
  k_linv<<<NH_ * OUT_, 128, 0, stream>>>(Lb, u_mean, Linv, rg);
  k_w<<<NH_ * OUT_, 128, 0, stream>>>(Linv, u_tril, Wb);
  k_q<<<NH_ * OUT_, 128, 0, stream>>>(Linv, Wb, Qb);
  k_main<<<dim3(B_ / 128, OUT_, NH_), 256, 0, stream>>>(
      x, zsc, z2, x2, sf2, Linv, Qb, rg, (float*)d_out);
}